// GCNLearnableModel_90031104458820
// MI455X (gfx1250) — compile-verified
//
#include <hip/hip_runtime.h>
#include <math.h>

// ---------------------------------------------------------------------------
// R-GCN (3 node types, 9 relations, 3 layers) for MI455X / gfx1250.
//   * Dense GEMMs use V_WMMA_F32_16X16X4_F32 (fp32, matches reference).
//   * A panel pair-swizzled in LDS -> single aligned ds_load_b64 per fragment.
//   * Weight panel staged with GLOBAL_LOAD_ASYNC_TO_LDS_B128 (ASYNCcnt path).
//   * Edge aggregation: per-relation transform T = (h*out_norm)@W via WMMA,
//     then scatter with global_atomic_add_f32 (T=12.8MB, acc=38.4MB, both
//     resident in the 192MB L2).
// ---------------------------------------------------------------------------

typedef __attribute__((ext_vector_type(2))) float v2f;
typedef __attribute__((ext_vector_type(8))) float v8f;
typedef __attribute__((ext_vector_type(4))) int v4i;

// Pointer-to-v4i in specific address spaces (async LDS builtin operand types).
typedef __attribute__((address_space(1))) v4i* gptr_v4i;
typedef __attribute__((address_space(3))) v4i* lptr_v4i;

constexpr int kN = 50000;   // nodes per type (multiple of 16)
constexpr int kE = 800000;  // edges per relation
constexpr int kREL = 9;

#if defined(__has_builtin)
#if __has_builtin(__builtin_amdgcn_global_load_async_to_lds_b128)
#define HAVE_ASYNC_LDS 1
#endif
#if __has_builtin(__builtin_amdgcn_s_wait_asynccnt)
#define HAVE_WAIT_ASYNC 1
#endif
#endif

// REL = [(0,1),(2,1),(2,0),(0,0),(1,2),(1,0),(0,0),(1,1),(2,2)]
__device__ __constant__ int d_DSTT[9] = {1, 1, 0, 0, 2, 0, 0, 1, 2};

// ---------------------------------------------------------------------------
__global__ void zero_k(float* __restrict__ p, long n) {
  long i = (long)blockIdx.x * blockDim.x + threadIdx.x;
  long stride = (long)gridDim.x * blockDim.x;
  for (; i < n; i += stride) p[i] = 0.f;
}

// Degrees over all 9 relations in one pass (src/dst are [9][E] flat).
__global__ void degree_k(const int* __restrict__ src, const int* __restrict__ dst,
                         float* __restrict__ outdeg, float* __restrict__ indeg) {
  long i = (long)blockIdx.x * blockDim.x + threadIdx.x;
  if (i >= (long)kREL * kE) return;
  int r = (int)(i / kE);
  atomicAdd(&outdeg[(size_t)r * kN + src[i]], 1.f);
  atomicAdd(&indeg[(size_t)r * kN + dst[i]], 1.f);
}

__global__ void normfin_k(float* __restrict__ p, long n) {
  long i = (long)blockIdx.x * blockDim.x + threadIdx.x;
  if (i < n) p[i] = rsqrtf(fmaxf(p[i], 1.f));
}

// ---------------------------------------------------------------------------
// WMMA fp32 GEMM: out[nrows,64] = (A[nrows,K] * scale[row]) @ W[K,64] + bias
// Block = 128 threads = 4 wave32s; each wave computes a 16x64 strip.
// nrows must be a multiple of 16 (50000 is).
template <int K>
__global__ __launch_bounds__(128) void gemm_wmma(
    const float* __restrict__ A, const float* __restrict__ scale,
    const float* __restrict__ W, const float* __restrict__ bias,
    float* __restrict__ out, int nrows) {
  // A panel pair-swizzled: elem(k, r) -> (k>>1)*128 + r*2 + (k&1).
  __shared__ float As[64 * K];
  __shared__ float Ws[K * 64];  // row-major [K][64]
  const int tid = threadIdx.x;
  const int rowblk = blockIdx.x * 64;

  // Stage weight panel (async DMA into LDS when available).
#if HAVE_ASYNC_LDS
  for (int idx = tid * 4; idx < K * 64; idx += 128 * 4) {
    __builtin_amdgcn_global_load_async_to_lds_b128(
        (gptr_v4i)(W + idx), (lptr_v4i)(&Ws[idx]), 0, 0);
  }
#else
  for (int idx = tid; idx < K * 64; idx += 128) Ws[idx] = W[idx];
#endif

  // Stage 64-row A panel (scaled) with pair swizzle (coalesced global reads).
  for (int idx = tid; idx < 64 * K; idx += 128) {
    int r = idx / K, k = idx - r * K;
    int grow = rowblk + r;
    float v = 0.f;
    if (grow < nrows) {
      v = A[(size_t)grow * K + k];
      if (scale) v *= scale[grow];
    }
    As[(k >> 1) * 128 + r * 2 + (k & 1)] = v;
  }

#if HAVE_ASYNC_LDS
#if HAVE_WAIT_ASYNC
  __builtin_amdgcn_s_wait_asynccnt(0);
#else
  asm volatile("s_wait_asynccnt 0x0" ::: "memory");
#endif
#endif
  __syncthreads();

  const int wave = tid >> 5;
  const int lane = tid & 31;
  const int rowbase = rowblk + wave * 16;
  if (rowbase >= nrows) return;  // wave-uniform; EXEC stays all-1s for WMMA

  const int half = lane >> 4;  // 0: K+0/1, C rows 0-7 ; 1: K+2/3, C rows 8-15
  const int l15 = lane & 15;
  const int koff = half * 2;

  v8f c[4];
  for (int ct = 0; ct < 4; ++ct) {
    float b = bias ? bias[ct * 16 + l15] : 0.f;
#pragma unroll
    for (int i = 0; i < 8; ++i) c[ct][i] = b;
  }

  // A fragment = one aligned b64 LDS read (pair-swizzled layout).
  // B fragment pairs come from adjacent K rows of row-major Ws.
  const int arow2 = (wave * 16 + l15) * 2;
  for (int k0 = 0; k0 < K; k0 += 4) {
    v2f a = *(const v2f*)&As[((k0 >> 1) + half) * 128 + arow2];
#pragma unroll
    for (int ct = 0; ct < 4; ++ct) {
      v2f b;
      b[0] = Ws[(k0 + koff) * 64 + ct * 16 + l15];
      b[1] = Ws[(k0 + koff + 1) * 64 + ct * 16 + l15];
      c[ct] = __builtin_amdgcn_wmma_f32_16x16x4_f32(
          false, a, false, b, (short)0, c[ct], false, false);
    }
  }

  // C/D layout: VGPR i -> M = half*8 + i, N = l15.
#pragma unroll
  for (int ct = 0; ct < 4; ++ct) {
#pragma unroll
    for (int i = 0; i < 8; ++i) {
      int m = half * 8 + i;
      out[(size_t)(rowbase + m) * 64 + ct * 16 + l15] = c[ct][i];
    }
  }
}

// ---------------------------------------------------------------------------
// Edge scatter: acc[dst][f] += T[src][f] * in_norm[dst].  64 threads per edge.
__global__ __launch_bounds__(256) void scatter_k(
    const float* __restrict__ T, const int* __restrict__ src,
    const int* __restrict__ dst, const float* __restrict__ innorm,
    float* __restrict__ acc) {
  long i = (long)blockIdx.x * 256 + threadIdx.x;
  if (i >= (long)kE * 64) return;
  long e = i >> 6;
  int f = (int)(i & 63);
  int s = src[e], d = dst[e];
  atomicAdd(&acc[(size_t)d * 64 + f], T[(size_t)s * 64 + f] * innorm[d]);
}

// Per-type folded conv bias: tbias[t][f] = sum_{r: dst(r)==t} conv_b[l,r,f]
__global__ void tbias_k(const float* __restrict__ convb_layer,
                        float* __restrict__ tbias) {
  int i = threadIdx.x;  // 192 threads
  int t = i >> 6, f = i & 63;
  float s = 0.f;
  for (int r = 0; r < kREL; ++r)
    if (d_DSTT[r] == t) s += convb_layer[r * 64 + f];
  tbias[i] = s;
}

// LayerNorm(acc + tbias) -> ReLU -> hout.  One wave32 per 64-feature row.
__global__ __launch_bounds__(256) void ln_relu_k(
    const float* __restrict__ acc, const float* __restrict__ tbias,
    const float* __restrict__ g, const float* __restrict__ b,
    float* __restrict__ hout, int n_per_type) {
  long row = ((long)blockIdx.x * blockDim.x + threadIdx.x) >> 5;
  int lane = threadIdx.x & 31;
  if (row >= 3L * n_per_type) return;
  int t = (int)(row / n_per_type);
  const float* x = acc + (size_t)row * 64;
  float x0 = x[lane] + tbias[t * 64 + lane];
  float x1 = x[lane + 32] + tbias[t * 64 + lane + 32];
  float s = x0 + x1;
  for (int o = 16; o > 0; o >>= 1) s += __shfl_xor(s, o, 32);
  float mu = s * (1.f / 64.f);
  float d0 = x0 - mu, d1 = x1 - mu;
  float v = d0 * d0 + d1 * d1;
  for (int o = 16; o > 0; o >>= 1) v += __shfl_xor(v, o, 32);
  float rstd = rsqrtf(v * (1.f / 64.f) + 1e-5f);
  float y0 = d0 * rstd * g[t * 64 + lane] + b[t * 64 + lane];
  float y1 = d1 * rstd * g[t * 64 + lane + 32] + b[t * 64 + lane + 32];
  float* o_ = hout + (size_t)row * 64;
  o_[lane] = fmaxf(y0, 0.f);
  o_[lane + 32] = fmaxf(y1, 0.f);
}

// Classifier: out[t,n,o] = h[t,n,:] . cls_W[t,:,o] + cls_b[t,o]
__global__ __launch_bounds__(256) void cls_k(
    const float* __restrict__ h, const float* __restrict__ clsW,
    const float* __restrict__ clsb, float* __restrict__ out, int n_per_type) {
  long i = (long)blockIdx.x * 256 + threadIdx.x;
  if (i >= 3L * n_per_type * 8) return;
  int t = (int)(i / ((long)n_per_type * 8));
  long rem = i - (long)t * n_per_type * 8;
  long n = rem >> 3;
  int o = (int)(rem & 7);
  const float* hr = h + ((size_t)t * n_per_type + n) * 64;
  const float* w = clsW + (size_t)t * 64 * 8 + o;
  float acc = clsb[t * 8 + o];
#pragma unroll
  for (int k = 0; k < 64; ++k) acc += hr[k] * w[k * 8];
  out[i] = acc;
}

// ---------------------------------------------------------------------------
extern "C" void kernel_launch(void* const* d_in, const int* in_sizes, int n_in,
                              void* d_out, int out_size, void* d_ws, size_t ws_size,
                              hipStream_t stream) {
  (void)in_sizes; (void)n_in; (void)out_size; (void)ws_size;
  const float* feat[3] = {(const float*)d_in[0], (const float*)d_in[1],
                          (const float*)d_in[2]};
  const float* emb_W = (const float*)d_in[3];   // [3,128,64]
  const float* emb_b = (const float*)d_in[4];   // [3,64]
  const float* conv_W = (const float*)d_in[5];  // [3,9,64,64]
  const float* conv_b = (const float*)d_in[6];  // [3,9,64]
  const float* ln_g = (const float*)d_in[7];    // [3,64]
  const float* ln_b = (const float*)d_in[8];    // [3,64]
  const float* cls_W = (const float*)d_in[9];   // [3,64,8]
  const float* cls_b = (const float*)d_in[10];  // [3,8]
  const int* src = (const int*)d_in[11];        // [9,E]
  const int* dst = (const int*)d_in[12];        // [9,E]

  // Workspace layout (floats), ~132 MB total.
  float* ws = (float*)d_ws;
  float* outnorm = ws;                           // [9,N]
  float* innorm = outnorm + (size_t)kREL * kN;   // [9,N]
  float* h0 = innorm + (size_t)kREL * kN;        // [3,N,64]
  float* h1 = h0 + (size_t)3 * kN * 64;          // [3,N,64]
  float* acc = h1 + (size_t)3 * kN * 64;         // [3,N,64]
  float* T = acc + (size_t)3 * kN * 64;          // [N,64]
  float* tbias = T + (size_t)kN * 64;            // [3,64]

  static const int SRCT[9] = {0, 2, 2, 0, 1, 1, 0, 1, 2};
  static const int DSTT[9] = {1, 1, 0, 0, 2, 0, 0, 1, 2};

  // --- Degree norms (shared across layers) ---
  zero_k<<<2048, 256, 0, stream>>>(outnorm, 2L * kREL * kN);
  degree_k<<<((long)kREL * kE + 255) / 256, 256, 0, stream>>>(src, dst, outnorm,
                                                              innorm);
  normfin_k<<<(2L * kREL * kN + 255) / 256, 256, 0, stream>>>(outnorm,
                                                              2L * kREL * kN);

  // --- Per-type embedding (WMMA fp32, K=128) ---
  const int gemm_blocks = (kN + 63) / 64;
  for (int t = 0; t < 3; ++t)
    gemm_wmma<128><<<gemm_blocks, 128, 0, stream>>>(
        feat[t], nullptr, emb_W + (size_t)t * 128 * 64, emb_b + t * 64,
        h0 + (size_t)t * kN * 64, kN);

  // --- 3 conv layers ---
  float* hcur = h0;
  float* hnext = h1;
  for (int l = 0; l < 3; ++l) {
    zero_k<<<2048, 256, 0, stream>>>(acc, 3L * kN * 64);
    for (int r = 0; r < kREL; ++r) {
      // T = (h[src_type] * out_norm_r) @ conv_W[l,r]   (WMMA fp32, K=64)
      gemm_wmma<64><<<gemm_blocks, 128, 0, stream>>>(
          hcur + (size_t)SRCT[r] * kN * 64, outnorm + (size_t)r * kN,
          conv_W + ((size_t)l * kREL + r) * 64 * 64, nullptr, T, kN);
      // acc[dst_type][d] += T[s] * in_norm_r[d]   (L2-resident atomics)
      scatter_k<<<((long)kE * 64 + 255) / 256, 256, 0, stream>>>(
          T, src + (size_t)r * kE, dst + (size_t)r * kE,
          innorm + (size_t)r * kN, acc + (size_t)DSTT[r] * kN * 64);
    }
    tbias_k<<<1, 192, 0, stream>>>(conv_b + (size_t)l * kREL * 64, tbias);
    ln_relu_k<<<(3L * kN * 32 + 255) / 256, 256, 0, stream>>>(acc, tbias, ln_g,
                                                              ln_b, hnext, kN);
    float* tmp = hcur; hcur = hnext; hnext = tmp;
  }

  // --- Classifier -> d_out [3,N,8] fp32 ---
  cls_k<<<(3L * kN * 8 + 255) / 256, 256, 0, stream>>>(hcur, cls_W, cls_b,
                                                       (float*)d_out, kN);
}